// AutoformerClassifier_23545010717117
// MI455X (gfx1250) — compile-verified
//
#include <hip/hip_runtime.h>
#include <hip/hip_bf16.h>
#include <math.h>

// ---------------- model constants ----------------
#define BB   64
#define TT   1024
#define CC   12
#define DD   512
#define HH   8
#define FFNN 2048
#define FCH  256
#define NCLS 5
#define TOPK 20
#define MCHUNK 8192   // token chunk for FFN hidden

typedef float v2f __attribute__((ext_vector_type(2)));
typedef float v8f __attribute__((ext_vector_type(8)));
// pointee type expected by the async-LDS builtin (gcc-style vector of 4 ints)
typedef int v4i_vs __attribute__((vector_size(16)));

// CDNA5 async global->LDS copy path (probe-gated; falls back to reg staging)
#if defined(__gfx1250__) && __has_builtin(__builtin_amdgcn_global_load_async_to_lds_b128)
#define HAVE_ASYNC_LDS 1
#else
#define HAVE_ASYNC_LDS 0
#endif

#if HAVE_ASYNC_LDS
#  if __has_builtin(__builtin_amdgcn_s_wait_asynccnt)
#    define WAIT_ASYNC() __builtin_amdgcn_s_wait_asynccnt(0)
#  else
#    define WAIT_ASYNC() asm volatile("s_wait_asynccnt 0" ::: "memory")
#  endif
#else
#  define WAIT_ASYNC() do {} while (0)
#endif

// =====================================================================
// 1) Mean-abs scaler: scale[b,c] = max(mean_t |x|, 1e-10)
// =====================================================================
__global__ __launch_bounds__(256) void scale_kernel(const float* __restrict__ x,
                                                    float* __restrict__ scale,
                                                    float* __restrict__ logscale) {
    int wave = threadIdx.x >> 5, lane = threadIdx.x & 31;
    int r = blockIdx.x * 8 + wave;
    if (r >= BB * CC) return;
    int b = r / CC, c = r % CC;
    const float* xp = x + (size_t)b * TT * CC + c;
    float s = 0.f;
    for (int t = lane; t < TT; t += 32) s += fabsf(xp[(size_t)t * CC]);
    #pragma unroll
    for (int m = 16; m > 0; m >>= 1) s += __shfl_xor(s, m, 32);
    if (lane == 0) {
        float sc = fmaxf(s * (1.f / TT), 1e-10f);
        scale[r] = sc;
        logscale[r] = logf(sc);
    }
}

// =====================================================================
// 2) ti = [x/scale (12) | zeros (12) | log(scale) (12)]
// =====================================================================
__global__ void build_ti_kernel(const float* __restrict__ x,
                                const float* __restrict__ scale,
                                const float* __restrict__ logscale,
                                float* __restrict__ ti) {
    size_t idx = (size_t)blockIdx.x * 256 + threadIdx.x;
    size_t total = (size_t)BB * TT * 36;
    if (idx >= total) return;
    int j = (int)(idx % 36);
    size_t bt = idx / 36;
    int b = (int)(bt / TT);
    float v;
    if (j < 12)      v = x[bt * CC + j] / scale[b * CC + j];
    else if (j < 24) v = 0.f;
    else             v = logscale[b * CC + (j - 24)];
    ti[idx] = v;
}

// =====================================================================
// 3) Embedding GEMV (K=36) + sinusoidal positional embedding
// =====================================================================
__global__ __launch_bounds__(256) void embed_kernel(const float* __restrict__ ti,
                                                    const float* __restrict__ emb_w,
                                                    float* __restrict__ h) {
    __shared__ float row[36];
    int bt = blockIdx.x;
    int t = bt & (TT - 1);
    if (threadIdx.x < 36) row[threadIdx.x] = ti[(size_t)bt * 36 + threadIdx.x];
    __syncthreads();
    #pragma unroll
    for (int rep = 0; rep < 2; ++rep) {
        int d = threadIdx.x + rep * 256;
        float acc = 0.f;
        #pragma unroll
        for (int j = 0; j < 36; ++j) acc += row[j] * emb_w[j * DD + d];
        int i = (d < 256) ? d : d - 256;
        float freq = __expf(-(float)(2 * i) * (9.21034037198f / 512.0f)); // ln(1e4)
        float ph = (float)t * freq;
        acc += (d < 256) ? sinf(ph) : cosf(ph);
        h[(size_t)bt * DD + d] = acc;
    }
}

// =====================================================================
// 4) LayerNorm over D=512, optional fused residual add (one wave / row)
// =====================================================================
__global__ __launch_bounds__(256) void layernorm_kernel(float* __restrict__ out,
                                                        const float* __restrict__ a,
                                                        const float* __restrict__ badd,
                                                        const float* __restrict__ g,
                                                        const float* __restrict__ beta,
                                                        int rows) {
    int wave = threadIdx.x >> 5, lane = threadIdx.x & 31;
    int row = blockIdx.x * 8 + wave;
    if (row >= rows) return;
    const float* pa = a + (size_t)row * DD;
    const float* pb = badd ? (badd + (size_t)row * DD) : nullptr;
    float vals[16];
    float s = 0.f;
    #pragma unroll
    for (int i = 0; i < 16; ++i) {
        int idx = lane + i * 32;
        float v = pa[idx];
        if (pb) v += pb[idx];
        vals[i] = v;
        s += v;
    }
    #pragma unroll
    for (int m = 16; m > 0; m >>= 1) s += __shfl_xor(s, m, 32);
    float mu = s * (1.f / DD);
    float vs = 0.f;
    #pragma unroll
    for (int i = 0; i < 16; ++i) { float dv = vals[i] - mu; vs += dv * dv; }
    #pragma unroll
    for (int m = 16; m > 0; m >>= 1) vs += __shfl_xor(vs, m, 32);
    float inv = rsqrtf(vs * (1.f / DD) + 1e-5f);
    float* po = out + (size_t)row * DD;
    #pragma unroll
    for (int i = 0; i < 16; ++i) {
        int idx = lane + i * 32;
        po[idx] = (vals[i] - mu) * inv * g[idx] + beta[idx];
    }
}

// =====================================================================
// 5) Tiled WMMA GEMM (fp32): Out[M,N] = A[M,K] @ Bw[K,N] + bias, opt GELU
//    block tile 128x64, 8 waves, wave tile 32x32 (2x2 of 16x16),
//    double-buffered LDS, A tile via GLOBAL_LOAD_ASYNC_TO_LDS_B128,
//    B tile register-staged & stored transposed (Bs[n][k]) so both A and
//    B fragments are contiguous ds_load_b64 reads.
//    Requires M%128==0, N%64==0, K%32==0. EXEC all-1s throughout.
// =====================================================================
__global__ __launch_bounds__(256) void gemm_wmma_kernel(const float* __restrict__ A,
                                                        const float* __restrict__ Bw,
                                                        const float* __restrict__ bias,
                                                        float* __restrict__ Out,
                                                        int M, int N, int K, int fuse_gelu) {
    __shared__ float As[2][128][36];   // [buf][m][k] (k-contiguous)
    __shared__ float Bs[2][64][36];    // [buf][n][k] (transposed, k-contiguous)
    int tid = threadIdx.x;
    int wave = tid >> 5, lane = tid & 31;
    int m0 = blockIdx.y * 128, n0 = blockIdx.x * 64;
    int wm = wave & 3, wn = wave >> 2;          // 4 x 2 wave grid
    int mbase = wm * 32, nbase = wn * 32;
    int m16 = lane & 15;
    int khalf = lane >> 4;                      // 0 / 1
    v8f acc[2][2] = {};

    // A tile: 128x32 = 1024 float4 -> 4 per thread
    int arow[4], acol[4];
    #pragma unroll
    for (int i = 0; i < 4; ++i) {
        int idx = tid + i * 256;
        arow[i] = idx >> 3;
        acol[i] = (idx & 7) * 4;
    }
    // B tile: 32x64 = 512 (n, k4) quads -> 2 per thread (4 k-strided b32 each)
    int bn[2], bk[2];
    #pragma unroll
    for (int i = 0; i < 2; ++i) {
        int idx = tid + i * 256;
        bn[i] = idx & 63;
        bk[i] = (idx >> 6) * 4;
    }

    float4 ra[4];
    float4 rb[2];

    auto prefetch = [&](int buf, int kc) {
        #pragma unroll
        for (int i = 0; i < 4; ++i) {
            const float* gp = A + (size_t)(m0 + arow[i]) * K + kc + acol[i];
#if HAVE_ASYNC_LDS
            __builtin_amdgcn_global_load_async_to_lds_b128(
                (__attribute__((address_space(1))) v4i_vs*)gp,
                (__attribute__((address_space(3))) v4i_vs*)&As[buf][arow[i]][acol[i]],
                0, 0);
#else
            ra[i] = *(const float4*)gp;
#endif
        }
        #pragma unroll
        for (int i = 0; i < 2; ++i) {
            const float* gb = Bw + (size_t)(kc + bk[i]) * N + n0 + bn[i];
            rb[i].x = gb[0 * (size_t)N];
            rb[i].y = gb[1 * (size_t)N];
            rb[i].z = gb[2 * (size_t)N];
            rb[i].w = gb[3 * (size_t)N];
        }
        (void)buf;
    };
    auto commit = [&](int buf) {
#if !HAVE_ASYNC_LDS
        #pragma unroll
        for (int i = 0; i < 4; ++i) *(float4*)(&As[buf][arow[i]][acol[i]]) = ra[i];
#endif
        #pragma unroll
        for (int i = 0; i < 2; ++i) *(float4*)(&Bs[buf][bn[i]][bk[i]]) = rb[i];
    };

    int nchunks = K / 32;
    prefetch(0, 0);
    commit(0);
    WAIT_ASYNC();
    __syncthreads();

    for (int c = 0; c < nchunks; ++c) {
        int cur = c & 1, nxt = cur ^ 1;
        if (c + 1 < nchunks) prefetch(nxt, (c + 1) * 32);   // overlap with compute
        #pragma unroll
        for (int kk = 0; kk < 8; ++kk) {
            int k0 = kk * 4 + khalf * 2;
            v2f a0, a1, b0, b1;
            a0.x = As[cur][mbase + m16][k0];        a0.y = As[cur][mbase + m16][k0 + 1];
            a1.x = As[cur][mbase + 16 + m16][k0];   a1.y = As[cur][mbase + 16 + m16][k0 + 1];
            b0.x = Bs[cur][nbase + m16][k0];        b0.y = Bs[cur][nbase + m16][k0 + 1];
            b1.x = Bs[cur][nbase + 16 + m16][k0];   b1.y = Bs[cur][nbase + 16 + m16][k0 + 1];
            acc[0][0] = __builtin_amdgcn_wmma_f32_16x16x4_f32(false, a0, false, b0, (short)0, acc[0][0], false, false);
            acc[0][1] = __builtin_amdgcn_wmma_f32_16x16x4_f32(false, a0, false, b1, (short)0, acc[0][1], false, false);
            acc[1][0] = __builtin_amdgcn_wmma_f32_16x16x4_f32(false, a1, false, b0, (short)0, acc[1][0], false, false);
            acc[1][1] = __builtin_amdgcn_wmma_f32_16x16x4_f32(false, a1, false, b1, (short)0, acc[1][1], false, false);
        }
        if (c + 1 < nchunks) commit(nxt);
        WAIT_ASYNC();
        __syncthreads();
    }

    // epilogue: D layout — VGPR r : lanes 0-15 -> M=r, lanes 16-31 -> M=r+8
    #pragma unroll
    for (int mi = 0; mi < 2; ++mi) {
        #pragma unroll
        for (int ni = 0; ni < 2; ++ni) {
            int col = n0 + nbase + ni * 16 + m16;
            float bv = bias[col];
            #pragma unroll
            for (int r = 0; r < 8; ++r) {
                int rowl = (lane < 16) ? r : (r + 8);
                int row = m0 + mbase + mi * 16 + rowl;
                float v = acc[mi][ni][r] + bv;
                if (fuse_gelu) v = 0.5f * v * (1.f + erff(v * 0.70710678118654752f));
                Out[(size_t)row * N + col] = v;
            }
        }
    }
    (void)M;
}

// =====================================================================
// 6) In-LDS 1024-point radix-2 FFT helper (256 threads)
// =====================================================================
__device__ inline float2 cmulf(float2 a, float2 b) {
    return make_float2(a.x * b.x - a.y * b.y, a.x * b.y + a.y * b.x);
}

__device__ void fft1024(float2* buf, int tid, float sign) {
    for (int i = tid; i < 1024; i += 256) {
        int j = (int)(__brev((unsigned)i) >> 22);
        if (i < j) { float2 tmp = buf[i]; buf[i] = buf[j]; buf[j] = tmp; }
    }
    __syncthreads();
    for (int len = 2; len <= 1024; len <<= 1) {
        int half = len >> 1;
        float ang = sign * 6.28318530717958647692f / (float)len;
        for (int i = tid; i < 512; i += 256) {
            int grp = i / half, pos = i - grp * half;
            int i1 = grp * len + pos, i2 = i1 + half;
            float sw, cw;
            __sincosf(ang * (float)pos, &sw, &cw);
            float2 w = make_float2(cw, sw);
            float2 u = buf[i1];
            float2 tv = cmulf(w, buf[i2]);
            buf[i1] = make_float2(u.x + tv.x, u.y + tv.y);
            buf[i2] = make_float2(u.x - tv.x, u.y - tv.y);
        }
        __syncthreads();
    }
}

// =====================================================================
// 7) Per-(b, 32 d's) forward FFTs of Q,K columns, accumulate Qf*conj(Kf)
//    into deterministic per-block partial sums. grid = (D/32, B)
// =====================================================================
__global__ __launch_bounds__(256) void corr_fft_kernel(const float* __restrict__ q,
                                                       const float* __restrict__ k,
                                                       float* __restrict__ part) {
    __shared__ float2 SQ[1024];
    __shared__ float2 SK[1024];
    __shared__ float2 SA[1024];
    int tid = threadIdx.x;
    int b = blockIdx.y;
    int g = blockIdx.x;
    int d0 = g * 32;
    const float* qb = q + (size_t)b * TT * DD;
    const float* kb = k + (size_t)b * TT * DD;
    for (int i = tid; i < 1024; i += 256) SA[i] = make_float2(0.f, 0.f);
    __syncthreads();
    for (int dd = 0; dd < 32; ++dd) {
        int d = d0 + dd;
        for (int i = tid; i < 1024; i += 256) {
            SQ[i] = make_float2(qb[(size_t)i * DD + d], 0.f);
            SK[i] = make_float2(kb[(size_t)i * DD + d], 0.f);
        }
        __syncthreads();
        fft1024(SQ, tid, -1.f);
        fft1024(SK, tid, -1.f);
        for (int i = tid; i < 1024; i += 256) {
            float2 a = SQ[i], c = SK[i];
            SA[i].x += a.x * c.x + a.y * c.y;   // Q * conj(K)
            SA[i].y += a.y * c.x - a.x * c.y;
        }
        __syncthreads();
    }
    float* pp = part + ((size_t)(b * 16 + g)) * 2048;
    for (int i = tid; i < 1024; i += 256) {
        pp[2 * i]     = SA[i].x;
        pp[2 * i + 1] = SA[i].y;
    }
}

// fixed-order reduction over the 16 partials per batch (deterministic)
__global__ void corr_reduce_kernel(const float* __restrict__ part, float* __restrict__ accum) {
    int idx = blockIdx.x * 256 + threadIdx.x;   // over B * 2048
    if (idx >= BB * 2048) return;
    int b = idx >> 11, r = idx & 2047;
    float s = 0.f;
    #pragma unroll
    for (int g = 0; g < 16; ++g) s += part[((size_t)(b * 16 + g) << 11) + r];
    accum[idx] = s;
}

// =====================================================================
// 8) Inverse FFT -> mean_corr, top-20 selection + softmax (one block / b)
// =====================================================================
__global__ __launch_bounds__(256) void corr_topk_kernel(const float* __restrict__ accum,
                                                        int* __restrict__ topidx,
                                                        float* __restrict__ topw) {
    __shared__ float2 S[1024];
    __shared__ float CV[1024];
    __shared__ float rmax[8];
    __shared__ int   rarg[8];
    __shared__ float selv[TOPK];
    __shared__ int   seli[TOPK];
    int tid = threadIdx.x;
    int b = blockIdx.x;
    const float2* ap = (const float2*)accum + (size_t)b * 1024;
    for (int i = tid; i < 1024; i += 256) S[i] = ap[i];
    __syncthreads();
    fft1024(S, tid, +1.f);                       // inverse (scale applied below)
    const float scl = 1.f / (1024.f * 512.f);    // 1/T (ifft) * 1/(H*hd) (mean)
    for (int i = tid; i < 1024; i += 256) CV[i] = S[i].x * scl;
    __syncthreads();
    for (int s = 0; s < TOPK; ++s) {
        float best = -1e30f; int bidx = 0;
        for (int i = tid; i < 1024; i += 256) {
            float v = CV[i];
            if (v > best) { best = v; bidx = i; }
        }
        #pragma unroll
        for (int m = 16; m > 0; m >>= 1) {
            float ov = __shfl_xor(best, m, 32);
            int   oi = __shfl_xor(bidx, m, 32);
            if (ov > best) { best = ov; bidx = oi; }
        }
        if ((tid & 31) == 0) { rmax[tid >> 5] = best; rarg[tid >> 5] = bidx; }
        __syncthreads();
        if (tid == 0) {
            for (int w = 1; w < 8; ++w)
                if (rmax[w] > best) { best = rmax[w]; bidx = rarg[w]; }
            selv[s] = best; seli[s] = bidx;
            CV[bidx] = -1e30f;
        }
        __syncthreads();
    }
    if (tid == 0) {
        float mx = selv[0];
        for (int s = 1; s < TOPK; ++s) mx = fmaxf(mx, selv[s]);
        float e[TOPK], sum = 0.f;
        for (int s = 0; s < TOPK; ++s) { e[s] = __expf(selv[s] - mx); sum += e[s]; }
        float inv = 1.f / sum;
        for (int s = 0; s < TOPK; ++s) {
            topw[b * TOPK + s] = e[s] * inv;
            topidx[b * TOPK + s] = seli[s];
        }
    }
}

// =====================================================================
// 9) Delay aggregation: out[b,t,:] = sum_i w_i * V[b,(t+delay_i)%T,:]
// =====================================================================
__global__ __launch_bounds__(256) void agg_kernel(const float* __restrict__ v,
                                                  const int* __restrict__ topidx,
                                                  const float* __restrict__ topw,
                                                  float* __restrict__ out) {
    __shared__ int   di[TOPK];
    __shared__ float dw[TOPK];
    int bt = blockIdx.x;
    int b = bt >> 10, t = bt & 1023;
    if (threadIdx.x < TOPK) {
        di[threadIdx.x] = topidx[b * TOPK + threadIdx.x];
        dw[threadIdx.x] = topw[b * TOPK + threadIdx.x];
    }
    __syncthreads();
    #pragma unroll
    for (int rep = 0; rep < 2; ++rep) {
        int d = threadIdx.x + rep * 256;
        float s = 0.f;
        for (int i = 0; i < TOPK; ++i) {
            int tt = (t + di[i]) & 1023;
            s += dw[i] * v[((size_t)b * TT + tt) * DD + d];
        }
        out[(size_t)bt * DD + d] = s;
    }
}

// =====================================================================
// 10) Series decomposition: seasonal = x - movavg_25(x, edge-replicate)
// =====================================================================
__global__ __launch_bounds__(256) void decomp_kernel(const float* __restrict__ in,
                                                     float* __restrict__ out) {
    int bt = blockIdx.x;
    int b = bt >> 10, t = bt & 1023;
    const float* xb = in + (size_t)b * TT * DD;
    #pragma unroll
    for (int rep = 0; rep < 2; ++rep) {
        int d = threadIdx.x + rep * 256;
        float s = 0.f;
        #pragma unroll
        for (int j = -12; j <= 12; ++j) {
            int tc = t + j;
            tc = tc < 0 ? 0 : (tc > TT - 1 ? TT - 1 : tc);
            s += xb[(size_t)tc * DD + d];
        }
        out[(size_t)bt * DD + d] = in[(size_t)bt * DD + d] - s * (1.f / 25.f);
    }
}

// =====================================================================
// 11) Mean pool over time, 12) classifier head
// =====================================================================
__global__ __launch_bounds__(256) void meanpool_kernel(const float* __restrict__ h,
                                                       float* __restrict__ feat) {
    int b = blockIdx.x;
    #pragma unroll
    for (int rep = 0; rep < 2; ++rep) {
        int d = threadIdx.x + rep * 256;
        float s = 0.f;
        for (int t = 0; t < TT; ++t) s += h[((size_t)b * TT + t) * DD + d];
        feat[b * DD + d] = s * (1.f / TT);
    }
}

__global__ __launch_bounds__(256) void head_kernel(const float* __restrict__ feat,
                                                   const float* __restrict__ w1,
                                                   const float* __restrict__ b1,
                                                   const float* __restrict__ w2,
                                                   const float* __restrict__ b2,
                                                   float* __restrict__ out) {
    __shared__ float f[DD];
    __shared__ float hid[FCH];
    int b = blockIdx.x;
    for (int i = threadIdx.x; i < DD; i += 256) f[i] = feat[b * DD + i];
    __syncthreads();
    int j = threadIdx.x;
    float s = b1[j];
    for (int k = 0; k < DD; ++k) s += f[k] * w1[k * FCH + j];
    hid[j] = fmaxf(s, 0.f);
    __syncthreads();
    if (j < NCLS) {
        float o = b2[j];
        for (int k = 0; k < FCH; ++k) o += hid[k] * w2[k * NCLS + j];
        out[b * NCLS + j] = o;
    }
}

// =====================================================================
// Host orchestration
// =====================================================================
extern "C" void kernel_launch(void* const* d_in, const int* in_sizes, int n_in,
                              void* d_out, int out_size, void* d_ws, size_t ws_size,
                              hipStream_t stream) {
    const float* x        = (const float*)d_in[0];
    const float* emb_w    = (const float*)d_in[1];
    const float* emb_ln_g = (const float*)d_in[2];
    const float* emb_ln_b = (const float*)d_in[3];
    const float* attn_w   = (const float*)d_in[4];   // (2,4,512,512)
    const float* attn_b   = (const float*)d_in[5];   // (2,4,512)
    const float* ln_g     = (const float*)d_in[6];   // (2,2,512)
    const float* ln_b     = (const float*)d_in[7];
    const float* fc1_w    = (const float*)d_in[8];   // (2,512,2048)
    const float* fc1_b    = (const float*)d_in[9];
    const float* fc2_w    = (const float*)d_in[10];  // (2,2048,512)
    const float* fc2_b    = (const float*)d_in[11];
    const float* hw1      = (const float*)d_in[12];
    const float* hb1      = (const float*)d_in[13];
    const float* hw2      = (const float*)d_in[14];
    const float* hb2      = (const float*)d_in[15];
    float* out = (float*)d_out;

    const size_t NTOK = (size_t)BB * TT;   // 65536

    // ---- workspace carve (float units, 4KB-aligned chunks) ----
    float* ws = (float*)d_ws;
    size_t off = 0;
    auto alloc = [&](size_t n) {
        float* p = ws + off;
        off += (n + 1023) & ~(size_t)1023;
        return p;
    };
    float* scale    = alloc(BB * CC);
    float* logscale = alloc(BB * CC);
    float* ti       = alloc(NTOK * 36);
    float* hA       = alloc(NTOK * DD);
    float* B1p      = alloc(NTOK * DD);
    float* B2p      = alloc(NTOK * DD);
    float* B3p      = alloc(NTOK * DD);
    float* hid      = alloc((size_t)MCHUNK * FFNN);
    float* part     = alloc((size_t)BB * 16 * 2048);
    float* accum    = alloc((size_t)BB * 2048);
    float* topw     = alloc(BB * TOPK);
    int*   topidx   = (int*)alloc(BB * TOPK);
    float* feat     = alloc(BB * DD);
    (void)ws_size; (void)in_sizes; (void)n_in; (void)out_size;

    // ---- scaler + feature build + embedding + LN ----
    scale_kernel<<<(BB * CC + 7) / 8, 256, 0, stream>>>(x, scale, logscale);
    size_t tot_ti = NTOK * 36;
    build_ti_kernel<<<(unsigned)((tot_ti + 255) / 256), 256, 0, stream>>>(x, scale, logscale, ti);
    embed_kernel<<<(unsigned)NTOK, 256, 0, stream>>>(ti, emb_w, hA);
    layernorm_kernel<<<(unsigned)(NTOK / 8), 256, 0, stream>>>(hA, hA, nullptr, emb_ln_g, emb_ln_b, (int)NTOK);

    dim3 gProj(DD / 64, (unsigned)(NTOK / 128));     // (8, 512)
    for (int l = 0; l < 2; ++l) {
        const float* w  = attn_w + (size_t)l * 4 * DD * DD;
        const float* bb = attn_b + (size_t)l * 4 * DD;
        // Q, K, V projections (WMMA)
        gemm_wmma_kernel<<<gProj, 256, 0, stream>>>(hA, w,                   bb,          B1p, (int)NTOK, DD, DD, 0);
        gemm_wmma_kernel<<<gProj, 256, 0, stream>>>(hA, w + (size_t)DD*DD,   bb + DD,     B2p, (int)NTOK, DD, DD, 0);
        gemm_wmma_kernel<<<gProj, 256, 0, stream>>>(hA, w + (size_t)2*DD*DD, bb + 2*DD,   B3p, (int)NTOK, DD, DD, 0);
        // autocorrelation: FFT accumulate -> reduce -> ifft + top-k + softmax
        corr_fft_kernel<<<dim3(DD / 32, BB), 256, 0, stream>>>(B1p, B2p, part);
        corr_reduce_kernel<<<(BB * 2048 + 255) / 256, 256, 0, stream>>>(part, accum);
        corr_topk_kernel<<<BB, 256, 0, stream>>>(accum, topidx, topw);
        // delay aggregation of V, then output projection
        agg_kernel<<<(unsigned)NTOK, 256, 0, stream>>>(B3p, topidx, topw, B1p);
        gemm_wmma_kernel<<<gProj, 256, 0, stream>>>(B1p, w + (size_t)3*DD*DD, bb + 3*DD, B2p, (int)NTOK, DD, DD, 0);
        // h = LN(res + attn) ; seasonal decomp
        layernorm_kernel<<<(unsigned)(NTOK / 8), 256, 0, stream>>>(hA, hA, B2p,
                                                                   ln_g + (size_t)l * 2 * DD,
                                                                   ln_b + (size_t)l * 2 * DD, (int)NTOK);
        decomp_kernel<<<(unsigned)NTOK, 256, 0, stream>>>(hA, B1p);
        // FFN (chunked over tokens): B2 = gelu(B1@fc1+b) @ fc2 + b
        for (int mc = 0; mc < (int)(NTOK / MCHUNK); ++mc) {
            const float* Ac = B1p + (size_t)mc * MCHUNK * DD;
            dim3 g1(FFNN / 64, MCHUNK / 128);        // (32, 64)
            gemm_wmma_kernel<<<g1, 256, 0, stream>>>(Ac, fc1_w + (size_t)l * DD * FFNN,
                                                     fc1_b + (size_t)l * FFNN, hid,
                                                     MCHUNK, FFNN, DD, 1);
            dim3 g2(DD / 64, MCHUNK / 128);          // (8, 64)
            gemm_wmma_kernel<<<g2, 256, 0, stream>>>(hid, fc2_w + (size_t)l * FFNN * DD,
                                                     fc2_b + (size_t)l * DD,
                                                     B2p + (size_t)mc * MCHUNK * DD,
                                                     MCHUNK, DD, FFNN, 0);
        }
        // h = LN(res + ffn) ; seasonal decomp -> hA for next layer
        layernorm_kernel<<<(unsigned)(NTOK / 8), 256, 0, stream>>>(B3p, B1p, B2p,
                                                                   ln_g + (size_t)l * 2 * DD + DD,
                                                                   ln_b + (size_t)l * 2 * DD + DD, (int)NTOK);
        decomp_kernel<<<(unsigned)NTOK, 256, 0, stream>>>(B3p, hA);
    }

    meanpool_kernel<<<BB, 256, 0, stream>>>(hA, feat);
    head_kernel<<<BB, 256, 0, stream>>>(feat, hw1, hb1, hw2, hb2, out);
}